// bce_disco_loss_38285338476743
// MI455X (gfx1250) — compile-verified
//
#include <hip/hip_runtime.h>
#include <hip/hip_bf16.h>

typedef __attribute__((ext_vector_type(2))) float v2f;
typedef __attribute__((ext_vector_type(8))) float v8f;

#define NMAX 8192
#define TILE_J 2048

// Async global->LDS path (CDNA5 GLOBAL_LOAD_ASYNC_TO_LDS_*, ASYNCcnt-tracked).
#if defined(__has_builtin)
#if __has_builtin(__builtin_amdgcn_global_load_async_to_lds_b128)
#define USE_ASYNC_LDS 1
#endif
#endif

#if defined(USE_ASYNC_LDS)
// From the probe diagnostic: param0 is non-const AS1 pointer to int __vector_size__(16).
typedef int v4i __attribute__((vector_size(16)));
typedef __attribute__((address_space(1))) v4i GV4;   // global
typedef __attribute__((address_space(3))) v4i LV4;   // LDS
__device__ __forceinline__ void async_b128(const float* gsrc, float* ldst) {
  __builtin_amdgcn_global_load_async_to_lds_b128(
      (GV4*)(v4i*)(void*)const_cast<float*>(gsrc),
      (LV4*)(v4i*)(void*)ldst, 0, 0);
}
__device__ __forceinline__ void wait_async() {
#if __has_builtin(__builtin_amdgcn_s_wait_asynccnt)
  __builtin_amdgcn_s_wait_asynccnt(0);
#else
  asm volatile("s_wait_asynccnt 0x0" ::: "memory");
#endif
}
#endif

// ws layout:
//   [0]              int n (background count)
//   [8..55]          6 doubles: bce_sum, win_sum, Sx, Sxx, Sy, Syy
//   [64 .. +32K)     float xb[8192]   (compacted background mass, zero-padded)
//   [64+32K .. +64K) float yb[8192]   (compacted background output, zero-padded)
//   [64+64K ..)      double partials[64*6]
struct __align__(16) Hdr { int n; int _pad; double s[6]; };

// ---------------- Kernel 1: BCE, moments, deterministic compaction ----------------
__global__ __launch_bounds__(1024)
void k1_prep(const float* __restrict__ out_p, const float* __restrict__ tgt_p,
             const float* __restrict__ mass_p, void* ws) {
  __shared__ int scn[1024];
  __shared__ double red[1024];
  const int tid = threadIdx.x;

  float m[8], o[8], t[8];
  int f[8];
  int cnt = 0;
#pragma unroll
  for (int k = 0; k < 8; ++k) {
    const int idx = tid * 8 + k;
    m[k] = mass_p[idx]; o[k] = out_p[idx]; t[k] = tgt_p[idx];
    f[k] = (t[k] == 0.0f) ? 1 : 0;
    cnt += f[k];
  }

  // inclusive Hillis-Steele scan over per-thread counts (deterministic compaction)
  scn[tid] = cnt; __syncthreads();
  for (int off = 1; off < 1024; off <<= 1) {
    const int v = scn[tid];
    const int add = (tid >= off) ? scn[tid - off] : 0;
    __syncthreads();
    scn[tid] = v + add;
    __syncthreads();
  }
  const int incl = scn[tid];
  const int n = scn[1023];
  int pos = incl - cnt;

  float* xb = (float*)((char*)ws + 64);
  float* yb = xb + NMAX;
#pragma unroll
  for (int k = 0; k < 8; ++k) {
    if (f[k]) { xb[pos] = m[k]; yb[pos] = o[k]; ++pos; }
  }
  // zero-pad so full fixed-size tile copies and out-of-range rows stay finite
  for (int idx = n + tid; idx < NMAX; idx += 1024) { xb[idx] = 0.0f; yb[idx] = 0.0f; }

  // BCE (mass window) + background moments
  double bce = 0.0, win = 0.0, sx = 0.0, sxx = 0.0, sy = 0.0, syy = 0.0;
#pragma unroll
  for (int k = 0; k < 8; ++k) {
    const float w = (m[k] > 120.0f && m[k] < 130.0f) ? 1.0f : 0.0f;
    const float lo = fmaxf(logf(o[k]), -100.0f);
    const float l1 = fmaxf(log1pf(-o[k]), -100.0f);
    const float b = -(t[k] * lo + (1.0f - t[k]) * l1);
    bce += (double)(b * w);
    win += (double)w;
    if (f[k]) {
      sx  += (double)m[k]; sxx += (double)m[k] * (double)m[k];
      sy  += (double)o[k]; syy += (double)o[k] * (double)o[k];
    }
  }

  auto block_red = [&](double v) -> double {
    red[tid] = v; __syncthreads();
    for (int s = 512; s > 0; s >>= 1) {
      if (tid < s) red[tid] += red[tid + s];
      __syncthreads();
    }
    const double r = red[0]; __syncthreads();
    return r;
  };
  const double rb = block_red(bce), rw = block_red(win);
  const double rsx = block_red(sx), rsxx = block_red(sxx);
  const double rsy = block_red(sy), rsyy = block_red(syy);
  if (tid == 0) {
    Hdr* h = (Hdr*)ws;
    h->n = n;
    h->s[0] = rb; h->s[1] = rw; h->s[2] = rsx; h->s[3] = rsxx; h->s[4] = rsy; h->s[5] = rsyy;
  }
}

// ---------------- Kernel 2: O(n^2) pair pass (WMMA row sums + VALU product sum) ----
__global__ __launch_bounds__(256)
void k2_pairs(void* ws) {
  __shared__ __align__(16) float xt[2][TILE_J];
  __shared__ __align__(16) float yt[2][TILE_J];
  __shared__ float sdx_l[128];
  __shared__ float sdy_l[128];
  __shared__ float sp_l[256];
  __shared__ double dred[256];

  const Hdr* h = (const Hdr*)ws;
  const int n = h->n;
  const float* xb = (const float*)((const char*)ws + 64);
  const float* yb = xb + NMAX;
  double* parts = (double*)((char*)ws + 64 + 2 * NMAX * 4);

  const int tid  = threadIdx.x;
  const int wave = tid >> 5;
  const int lane = tid & 31;
  const int b    = blockIdx.x;
  const int i0b  = b * 128;

  if (i0b >= n) {              // whole block out of range: emit zero partials
    if (tid < 6) parts[b * 6 + tid] = 0.0;
    return;
  }

  // Fixed full-tile copy (sources are zero-padded to NMAX; TILE_J divides NMAX).
  auto tile_issue = [&](int buf, int tb) {
#if defined(USE_ASYNC_LDS)
    const int o0 = tid * 4, o1 = (tid + 256) * 4;   // 2 x b128 per thread per array
    async_b128(xb + tb + o0, &xt[buf][o0]);
    async_b128(xb + tb + o1, &xt[buf][o1]);
    async_b128(yb + tb + o0, &yt[buf][o0]);
    async_b128(yb + tb + o1, &yt[buf][o1]);
#else
    for (int idx = tid; idx < TILE_J; idx += 256) {
      xt[buf][idx] = xb[tb + idx];
      yt[buf][idx] = yb[tb + idx];
    }
#endif
  };
  auto tile_wait = [&]() {
#if defined(USE_ASYNC_LDS)
    wait_async();
#endif
    __syncthreads();
  };

  // WMMA A-layout (16x4 f32): lanes 0-15 hold K0,K1 of row=lane; lanes 16-31 hold K2,K3.
  const int row  = lane & 15;
  const int i    = i0b + wave * 16 + row;   // this lane's i-row (always < 8192)
  const float xi = xb[i];
  const float yi = yb[i];
  const int koff = (lane < 16) ? 0 : 2;

  v8f cx = {0.f,0.f,0.f,0.f,0.f,0.f,0.f,0.f};   // row sums of |dx| (replicated per column)
  v8f cy = {0.f,0.f,0.f,0.f,0.f,0.f,0.f,0.f};   // row sums of |dy|
  const v2f bones = {1.0f, 1.0f};               // B = ones(4x16); layout-independent
  float sp = 0.0f;                              // per-lane sum of |dx|*|dy|
  float tdx = 0.0f, tdy = 0.0f;                 // tail row-sum contributions (lanes 0-15)

  auto do_chunk = [&](const float* __restrict__ xtb, const float* __restrict__ ytb, int jc) {
    const float xj0 = xtb[jc + koff], xj1 = xtb[jc + koff + 1];
    const float yj0 = ytb[jc + koff], yj1 = ytb[jc + koff + 1];
    const float ax0 = fabsf(xi - xj0), ax1 = fabsf(xi - xj1);
    const float ay0 = fabsf(yi - yj0), ay1 = fabsf(yi - yj1);
    const v2f ax = {ax0, ax1};
    const v2f ay = {ay0, ay1};
    cx = __builtin_amdgcn_wmma_f32_16x16x4_f32(false, ax, false, bones, (short)0, cx, false, false);
    cy = __builtin_amdgcn_wmma_f32_16x16x4_f32(false, ay, false, bones, (short)0, cy, false, false);
    sp = fmaf(ax0, ay0, sp);
    sp = fmaf(ax1, ay1, sp);
  };

  // Prime the pipeline: tile 0 into buffer 0.
  tile_issue(0, 0);
  tile_wait();

  const int nt = (n + TILE_J - 1) / TILE_J;
  for (int t = 0; t < nt; ++t) {
    const int buf = t & 1;
    if (t + 1 < nt) tile_issue(buf ^ 1, (t + 1) * TILE_J);   // overlap next tile's DMA

    const int tb   = t * TILE_J;
    const int tlen = min(TILE_J, n - tb);
    const float* __restrict__ xtb = xt[buf];
    const float* __restrict__ ytb = yt[buf];

    const int t4 = tlen & ~3;
    const int t8 = tlen & ~7;
    for (int jc = 0; jc < t8; jc += 8) {       // unrolled x2: 4 WMMA / iteration
      do_chunk(xtb, ytb, jc);
      do_chunk(xtb, ytb, jc + 4);
    }
    if (t4 > t8) do_chunk(xtb, ytb, t8);
    if (lane < 16) {                           // <4 leftover j's: scalar, keeps EXEC=all-1s
      for (int j = t4; j < tlen; ++j) {
        const float dx = fabsf(xi - xtb[j]);
        const float dy = fabsf(yi - ytb[j]);
        tdx += dx; tdy += dy;
        sp = fmaf(dx, dy, sp);
      }
    }
    tile_wait();   // next tile's DMA done + all waves finished reading this buffer
  }

  // Extract row sums: D VGPR v holds row v (lanes 0-15) / row v+8 (lanes 16-31).
  if (lane == 0 || lane == 16) {
    const int base = wave * 16 + ((lane == 16) ? 8 : 0);
#pragma unroll
    for (int v = 0; v < 8; ++v) { sdx_l[base + v] = cx[v]; sdy_l[base + v] = cy[v]; }
  }
  sp_l[tid] = sp;
  __syncthreads();

  double pP = 0.0, pRx = 0.0, pRy = 0.0, pRxx = 0.0, pRyy = 0.0, pRxy = 0.0;
  if (lane < 16) {
    const int il = wave * 16 + lane;
    const int gi = i0b + il;
    if (gi < n) {
      const double rx = (double)sdx_l[il] + (double)tdx;
      const double ry = (double)sdy_l[il] + (double)tdy;
      const double p  = (double)sp_l[wave * 32 + lane] + (double)sp_l[wave * 32 + lane + 16];
      pP = p; pRx = rx; pRy = ry;
      pRxx = rx * rx; pRyy = ry * ry; pRxy = rx * ry;
    }
  }

  auto block_red = [&](double v) -> double {
    dred[tid] = v; __syncthreads();
    for (int s = 128; s > 0; s >>= 1) {
      if (tid < s) dred[tid] += dred[tid + s];
      __syncthreads();
    }
    const double r = dred[0]; __syncthreads();
    return r;
  };
  const double tP = block_red(pP), tRx = block_red(pRx), tRy = block_red(pRy);
  const double tRxx = block_red(pRxx), tRyy = block_red(pRyy), tRxy = block_red(pRxy);
  if (tid == 0) {
    double* p = &parts[b * 6];
    p[0] = tP; p[1] = tRx; p[2] = tRy; p[3] = tRxx; p[4] = tRyy; p[5] = tRxy;
  }
}

// ---------------- Kernel 3: f64 combination + final scalar ----------------
__global__ __launch_bounds__(64)
void k3_final(void* ws, float* out) {
  __shared__ double dred[64];
  const int tid = threadIdx.x;
  const double* parts = (const double*)((const char*)ws + 64 + 2 * NMAX * 4);

  double v[6];
#pragma unroll
  for (int q = 0; q < 6; ++q) v[q] = parts[tid * 6 + q];

  double tot[6];
#pragma unroll
  for (int q = 0; q < 6; ++q) {
    dred[tid] = v[q]; __syncthreads();
    for (int s = 32; s > 0; s >>= 1) {
      if (tid < s) dred[tid] += dred[tid + s];
      __syncthreads();
    }
    tot[q] = dred[0]; __syncthreads();
  }

  if (tid == 0) {
    const Hdr* h = (const Hdr*)ws;
    const double nn  = (double)h->n;
    const double bce = h->s[0] / h->s[1];
    const double Sx = h->s[2], Sxx = h->s[3], Sy = h->s[4], Syy = h->s[5];
    const double P = tot[0], Rx = tot[1], Ry = tot[2];
    const double Rxx = tot[3], Ryy = tot[4], Rxy = tot[5];

    const double S_AB = P - 2.0 * Rxy / nn + (Rx * Ry) / (nn * nn);
    const double Paa  = 2.0 * nn * Sxx - 2.0 * Sx * Sx;   // sum (x_i-x_j)^2 over bkg
    const double Pbb  = 2.0 * nn * Syy - 2.0 * Sy * Sy;
    const double S_AA = Paa - 2.0 * Rxx / nn + (Rx * Rx) / (nn * nn);
    const double S_BB = Pbb - 2.0 * Ryy / nn + (Ry * Ry) / (nn * nn);
    const double disco = S_AB / sqrt(S_AA * S_BB);

    out[0] = (float)(bce + 10.0 * disco);
  }
}

extern "C" void kernel_launch(void* const* d_in, const int* in_sizes, int n_in,
                              void* d_out, int out_size, void* d_ws, size_t ws_size,
                              hipStream_t stream) {
  const float* out_p  = (const float*)d_in[0];  // output
  const float* tgt_p  = (const float*)d_in[1];  // target
  const float* mass_p = (const float*)d_in[2];  // mass
  (void)in_sizes; (void)n_in; (void)out_size; (void)ws_size;

  k1_prep <<<1, 1024, 0, stream>>>(out_p, tgt_p, mass_p, d_ws);
  k2_pairs<<<64, 256, 0, stream>>>(d_ws);
  k3_final<<<1, 64, 0, stream>>>(d_ws, (float*)d_out);
}